// UnbiasedPatchNCELoss_28303834480972
// MI455X (gfx1250) — compile-verified
//
#include <hip/hip_runtime.h>
#include <hip/hip_bf16.h>

#define BATCH   4
#define NPATCH  4096
#define DIM     256
#define MAXC    5
#define INV_T   (1.0f / 0.07f)
#define NEG_FILL (-10.0f)

#define ROWS_WG 64      // rows of src per workgroup (4 waves * 16)
#define TCOL    32      // tgt columns per LDS tile (double buffered)
#define NTILES  (NPATCH / TCOL)
#define LDSPAD  260     // 256 + 4 float pad: row skew of 4 banks, float4-aligned

typedef __attribute__((ext_vector_type(2))) float v2f;
typedef __attribute__((ext_vector_type(8))) float v8f;

#define DEV_INF (__builtin_inff())

// ---------------------------------------------------------------------------
// Tensor Data Mover: DMA a [rows x 256] fp32 tile (row stride 256) from global
// into LDS with a 4-DWORD pad after every 256 DWORDs (-> LDS row stride 260).
// D# layout per CDNA5 ISA 8.3/8.4.  This toolchain uses the 6-arg builtin:
//   (uint32x4 g0, int32x8 g1, int32x4 g2, int32x4 g3, int32x8, i32 cpol)
// ---------------------------------------------------------------------------
#if __has_builtin(__builtin_amdgcn_tensor_load_to_lds)
#define HAVE_TDM 1
typedef __attribute__((ext_vector_type(4))) unsigned int v4u;
typedef __attribute__((ext_vector_type(8))) int v8i;
typedef __attribute__((ext_vector_type(4))) int v4i;

__device__ __forceinline__ void tdm_load_tile_f32(const float* gsrc,
                                                  void* lds_dst, int rows) {
  unsigned long long ga = (unsigned long long)(size_t)gsrc;
  unsigned lds = (unsigned)(size_t)lds_dst;      // low 32 bits = LDS offset
  v4u g0;
  g0.x = 1u;                                     // count=1 (valid descriptor)
  g0.y = lds;                                    // lds_addr
  g0.z = (unsigned)(ga & 0xFFFFFFFFull);         // global_addr[31:0]
  g0.w = (unsigned)((ga >> 32) & 0x01FFFFFFull)  // global_addr[56:32]
         | (2u << 30);                           // type = 2 ("image")
  v8i g1;
  g1[0] = (int)((2u << 16)       // data_size = 4 bytes
              | (1u << 20)       // pad_enable
              | (7u << 22)       // pad_interval: 256 DWORDs
              | (3u << 25));     // pad_amount:   4 DWORDs
  g1[1] = (int)(256u << 16);     // tensor_dim0 = 256 elements
  g1[2] = (int)(4096u << 16);    // tensor_dim1 = 4096 lines
  g1[3] = (int)(256u << 16);     // tile_dim0 = 256
  g1[4] = rows;                  // tile_dim1 = rows
  g1[5] = 256;                   // tensor_dim0_stride = 256 elements
  g1[6] = 0;
  g1[7] = 0;
  v4i z4 = {0, 0, 0, 0};
  v8i z8 = {0, 0, 0, 0, 0, 0, 0, 0};
  __builtin_amdgcn_tensor_load_to_lds(g0, g1, z4, z4, z8, 0);
}
#endif

// ---------------------------------------------------------------------------
// Kernel 1: positive logits  pos[n] = dot(src[n], tgt[n]) / T
// ---------------------------------------------------------------------------
__global__ __launch_bounds__(256) void nce_pos_kernel(
    const float* __restrict__ src, const float* __restrict__ tgt,
    float* __restrict__ pos) {
  const int wave = threadIdx.x >> 5;
  const int lane = threadIdx.x & 31;
  const int row  = blockIdx.x * 8 + wave;
  const float* a = src + (size_t)row * DIM + lane * 8;
  const float* b = tgt + (size_t)row * DIM + lane * 8;
  float4 a0 = ((const float4*)a)[0], a1 = ((const float4*)a)[1];
  float4 b0 = ((const float4*)b)[0], b1 = ((const float4*)b)[1];
  float s = a0.x * b0.x + a0.y * b0.y + a0.z * b0.z + a0.w * b0.w +
            a1.x * b1.x + a1.y * b1.y + a1.z * b1.z + a1.w * b1.w;
#pragma unroll
  for (int m = 1; m <= 16; m <<= 1) s += __shfl_xor(s, m, 32);
  if (lane == 0) pos[row] = s * INV_T;
}

// ---------------------------------------------------------------------------
// Kernel 2: fused WMMA logits + per-lane online masked min / logsumexp.
// Cross-lane butterfly happens exactly once per row, after the column loop.
// Block: 128 threads (4 wave32), 16 rows per wave, TDM double-buffered tiles.
// ---------------------------------------------------------------------------
__global__ __launch_bounds__(128) void nce_logits_kernel(
    const float* __restrict__ src, const float* __restrict__ tgt,
    const int* __restrict__ cls,
    float* __restrict__ rowmin_out, float* __restrict__ lseneg_out) {
  __shared__ float sS[ROWS_WG][LDSPAD];
  __shared__ float sT[2][TCOL][LDSPAD];
  __shared__ int   sC[2][TCOL];

  const int b    = blockIdx.y;
  const int rb   = blockIdx.x * ROWS_WG;
  const int tid  = threadIdx.x;
  const int wave = tid >> 5;
  const int lane = tid & 31;
  const int m    = lane & 15;   // M (A) / N (B,C,D) index within tile
  const int hi   = lane >> 4;   // half-wave: K pair (A/B), M+8 (C/D)
  const int koff = hi * 2;
  const size_t base = (size_t)b * NPATCH;

  // ---- prologue: stage src rows + first tgt tile ----
#if HAVE_TDM
  if (wave == 0) {
    tdm_load_tile_f32(src + (base + rb) * DIM, &sS[0][0], ROWS_WG);
    tdm_load_tile_f32(tgt + base * DIM, &sT[0][0][0], TCOL);
  }
#else
  for (int idx = tid; idx < ROWS_WG * (DIM / 4); idx += 128) {
    int r = idx >> 6, q = idx & 63;
    float4 v = *(const float4*)(src + (base + rb + r) * DIM + q * 4);
    *(float4*)&sS[r][q * 4] = v;
  }
  for (int idx = tid; idx < TCOL * (DIM / 4); idx += 128) {
    int r = idx >> 6, q = idx & 63;
    float4 v = *(const float4*)(tgt + (base + r) * DIM + q * 4);
    *(float4*)&sT[0][r][q * 4] = v;
  }
#endif
  if (tid < TCOL) sC[0][tid] = cls[base + tid];

  const int rowbase = rb + wave * 16;
  int   rcls[8];
  float rmin[8], rmax[8], rsum[8];   // per-lane partial state (cols == m mod 16)
#pragma unroll
  for (int r = 0; r < 8; ++r) {
    rcls[r] = cls[base + rowbase + r + 8 * hi];
    rmin[r] = DEV_INF;
    rmax[r] = -DEV_INF;
    rsum[r] = 0.f;
  }

#if HAVE_TDM
  if (wave == 0) __builtin_amdgcn_s_wait_tensorcnt(0);
#endif
  __syncthreads();

  for (int t = 0; t < NTILES; ++t) {
    const int cur = t & 1;
    if (t + 1 < NTILES) {
      const int nxt = cur ^ 1;
#if HAVE_TDM
      if (wave == 0)
        tdm_load_tile_f32(tgt + (base + (size_t)(t + 1) * TCOL) * DIM,
                          &sT[nxt][0][0], TCOL);
#else
      for (int idx = tid; idx < TCOL * (DIM / 4); idx += 128) {
        int r = idx >> 6, q = idx & 63;
        float4 v =
            *(const float4*)(tgt + (base + (t + 1) * TCOL + r) * DIM + q * 4);
        *(float4*)&sT[nxt][r][q * 4] = v;
      }
#endif
      if (tid < TCOL) sC[nxt][tid] = cls[base + (t + 1) * TCOL + tid];
    }

    // ---- WMMA: two 16x16 logit tiles, exact fp32, K = 256 ----
    v8f acc[2];
#pragma unroll
    for (int j = 0; j < 2; ++j)
#pragma unroll
      for (int r = 0; r < 8; ++r) acc[j][r] = 0.f;

#pragma unroll 8
    for (int k = 0; k < DIM; k += 4) {
      v2f a = *(const v2f*)&sS[wave * 16 + m][k + koff];
#pragma unroll
      for (int j = 0; j < 2; ++j) {
        v2f bb = *(const v2f*)&sT[cur][j * 16 + m][k + koff];
        acc[j] = __builtin_amdgcn_wmma_f32_16x16x4_f32(
            false, a, false, bb, (short)0, acc[j], false, false);
      }
    }

    // ---- per-lane online reductions (no cross-lane traffic) ----
    const int cc0 = sC[cur][m];
    const int cc1 = sC[cur][16 + m];
#pragma unroll
    for (int r = 0; r < 8; ++r) {
      const int rc = rcls[r];
      float v0 = acc[0][r], v1 = acc[1][r];
      bool s0 = (cc0 == rc), s1 = (cc1 == rc);
      rmin[r] = fminf(rmin[r],
                      fminf(s0 ? DEV_INF : v0, s1 ? DEV_INF : v1));
      float y0 = s0 ? -DEV_INF : v0 * INV_T;
      float y1 = s1 ? -DEV_INF : v1 * INV_T;
      float nm = fmaxf(rmax[r], fmaxf(y0, y1));
      float nms = (nm == -DEV_INF) ? 0.f : nm;   // exp(-inf - 0) = 0 below
      rsum[r] = rsum[r] * __expf(rmax[r] - nms) + __expf(y0 - nms) +
                __expf(y1 - nms);
      rmax[r] = nm;
    }

    __syncthreads();            // all reads of sT[cur] complete
#if HAVE_TDM
    if (wave == 0 && t + 1 < NTILES) __builtin_amdgcn_s_wait_tensorcnt(0);
    __syncthreads();            // next buffer visible to every wave
#endif
  }

  // ---- final cross-lane merge: one butterfly per row ----
#pragma unroll
  for (int r = 0; r < 8; ++r) {
    float mn = rmin[r];
    float mx = rmax[r];
    float sm = rsum[r];
#pragma unroll
    for (int msk = 1; msk <= 8; msk <<= 1) {
      mn = fminf(mn, __shfl_xor(mn, msk, 32));
      float om = __shfl_xor(mx, msk, 32);
      float os = __shfl_xor(sm, msk, 32);
      float M  = fmaxf(mx, om);
      float Ms = (M == -DEV_INF) ? 0.f : M;
      sm = sm * __expf(mx - Ms) + os * __expf(om - Ms);
      mx = M;
    }
    if (m == 0) {               // lanes 0 / 16 hold rows r / r+8
      size_t row = base + rowbase + r + 8 * hi;
      rowmin_out[row] = mn;
      lseneg_out[row] = (sm > 0.f) ? mx + __logf(sm) : -DEV_INF;
    }
  }
}

// ---------------------------------------------------------------------------
// Kernel 3: per-class min of rowmin and class counts (one block per batch).
// ---------------------------------------------------------------------------
__device__ __forceinline__ unsigned fenc(float f) {
  unsigned u = __float_as_uint(f);
  return (u & 0x80000000u) ? ~u : (u | 0x80000000u);
}
__device__ __forceinline__ float fdec(unsigned e) {
  unsigned u = (e & 0x80000000u) ? (e ^ 0x80000000u) : ~e;
  return __uint_as_float(u);
}

__global__ __launch_bounds__(256) void nce_classred_kernel(
    const int* __restrict__ cls, const float* __restrict__ rmin_in,
    float* __restrict__ smallest, int* __restrict__ counts) {
  __shared__ unsigned smin[MAXC];
  __shared__ int      scnt[MAXC];
  const int b = blockIdx.x, t = threadIdx.x;
  if (t < MAXC) { smin[t] = 0xFFFFFFFFu; scnt[t] = 0; }
  __syncthreads();
  const size_t base = (size_t)b * NPATCH;
  for (int n = t; n < NPATCH; n += 256) {
    int c = cls[base + n];
    atomicMin(&smin[c], fenc(rmin_in[base + n]));
    atomicAdd(&scnt[c], 1);
  }
  __syncthreads();
  if (t < MAXC) {
    smallest[b * MAXC + t] = fminf(fdec(smin[t]), NEG_FILL) * INV_T;  // /T
    counts[b * MAXC + t]   = scnt[t];
  }
}

// ---------------------------------------------------------------------------
// Kernel 4: per-row loss = LSE([pos, lse_neg, pad]) - pos
// ---------------------------------------------------------------------------
__global__ __launch_bounds__(256) void nce_loss_kernel(
    const int* __restrict__ cls, const float* __restrict__ pos,
    const float* __restrict__ lseneg, const float* __restrict__ smallest,
    const int* __restrict__ counts, float* __restrict__ loss) {
  int n = blockIdx.x * 256 + threadIdx.x;
  int b = n / NPATCH;
  int c = cls[n];
  float p   = pos[n];
  float ln  = lseneg[n];
  float sm  = smallest[b * MAXC + c];
  int   np  = counts[b * MAXC + c] - 1;
  float pad = (np > 0) ? (__logf((float)np) + sm) : -DEV_INF;
  float mx  = fmaxf(p, fmaxf(ln, pad));
  float s   = __expf(p - mx)
            + ((ln  > -DEV_INF) ? __expf(ln  - mx) : 0.f)
            + ((pad > -DEV_INF) ? __expf(pad - mx) : 0.f);
  loss[n] = mx + __logf(s) - p;
}

// ---------------------------------------------------------------------------
// Kernel 5: stable counting-sort reorder per batch (== stable argsort(cls)).
// ---------------------------------------------------------------------------
__global__ __launch_bounds__(256) void nce_reorder_kernel(
    const int* __restrict__ cls, const float* __restrict__ loss,
    float* __restrict__ out) {
  __shared__ int cnt[MAXC][256];
  __shared__ int tot[MAXC];
  __shared__ int cbase[MAXC];
  const int b = blockIdx.x, t = threadIdx.x;
  const size_t base = (size_t)b * NPATCH;
  const int start = t * 16;

  int lc[MAXC] = {0, 0, 0, 0, 0};
  for (int i = 0; i < 16; ++i) lc[cls[base + start + i]]++;
  for (int c = 0; c < MAXC; ++c) cnt[c][t] = lc[c];
  __syncthreads();
  if (t < MAXC) {
    int s = 0;
    for (int j = 0; j < 256; ++j) { int v = cnt[t][j]; cnt[t][j] = s; s += v; }
    tot[t] = s;
  }
  __syncthreads();
  if (t == 0) {
    int s = 0;
    for (int c = 0; c < MAXC; ++c) { cbase[c] = s; s += tot[c]; }
  }
  __syncthreads();
  int run[MAXC];
  for (int c = 0; c < MAXC; ++c) run[c] = cbase[c] + cnt[c][t];
  for (int i = 0; i < 16; ++i) {
    int c = cls[base + start + i];
    out[base + run[c]++] = loss[base + start + i];
  }
}

// ---------------------------------------------------------------------------
extern "C" void kernel_launch(void* const* d_in, const int* in_sizes, int n_in,
                              void* d_out, int out_size, void* d_ws, size_t ws_size,
                              hipStream_t stream) {
  const float* src = (const float*)d_in[0];
  const float* tgt = (const float*)d_in[1];
  const int*   cls = (const int*)d_in[2];
  float* out = (float*)d_out;
  const int N = BATCH * NPATCH;

  float* pos      = (float*)d_ws;                     // N
  float* rmin     = pos + N;                          // N
  float* lse      = rmin + N;                         // N
  float* loss     = lse + N;                          // N
  float* smallest = loss + N;                         // B*MAXC
  int*   counts   = (int*)(smallest + BATCH * MAXC);  // B*MAXC

  nce_pos_kernel<<<N / 8, 256, 0, stream>>>(src, tgt, pos);

  dim3 g1(NPATCH / ROWS_WG, BATCH);
  nce_logits_kernel<<<g1, 128, 0, stream>>>(src, tgt, cls, rmin, lse);

  nce_classred_kernel<<<BATCH, 256, 0, stream>>>(cls, rmin, smallest, counts);

  nce_loss_kernel<<<N / 256, 256, 0, stream>>>(cls, pos, lse, smallest, counts, loss);

  nce_reorder_kernel<<<BATCH, 256, 0, stream>>>(cls, loss, out);
}